// BackBoneDistanceEmbedding_32736240730463
// MI455X (gfx1250) — compile-verified
//
#include <hip/hip_runtime.h>
#include <hip/hip_bf16.h>
#include <math.h>

// ---------------------------------------------------------------------------
// BackBoneDistanceEmbedding for MI455X (gfx1250, wave32).
// Heavy phase: N x N pairwise d^2 via V_WMMA_F32_16X16X4_F32 Gram tiles
// (d^2 = |pi|^2 + |pj|^2 - 2 pi.pj), fused with per-row top-20 selection.
// ---------------------------------------------------------------------------

typedef __attribute__((ext_vector_type(2))) float v2f;
typedef __attribute__((ext_vector_type(8))) float v8f;

#define NUM_N 20
#define PED   16

// 10000^(-t/8) = 10^(-t/2), t = 0..7
__device__ __constant__ float kFreq[8] = {
    1.0f, 0.31622776601683794f, 0.1f, 0.031622776601683794f,
    0.01f, 0.0031622776601683794f, 0.001f, 0.00031622776601683794f};

// Sorted (ascending) top-20 insertion; arrays stay in VGPRs (constant indices,
// fully unrolled). Fail-fast guard lowers to v_cmpx + exec-skip of the chain.
__device__ __forceinline__ void insert20(float (&bd)[NUM_N], int (&bi)[NUM_N],
                                         float d, int idx) {
  if (d < bd[NUM_N - 1]) {
    float cd = d;
    int ci = idx;
#pragma unroll
    for (int k = 0; k < NUM_N; ++k) {
      if (cd < bd[k]) {
        float tf = bd[k]; bd[k] = cd; cd = tf;
        int   ti = bi[k]; bi[k] = ci; ci = ti;
      }
    }
  }
}

// ---------------------------------------------------------------------------
// Kernel A: positions, |p|^2, padded float4 positions, pos3d embedding,
// full_edge_index row 1 (repeat(arange(N), 20)).
// ---------------------------------------------------------------------------
__global__ void prep_kernel(const float* __restrict__ aff,
                            float* __restrict__ pos_out,
                            float* __restrict__ emb_out,
                            float4* __restrict__ P4,
                            float* __restrict__ rsq,
                            int* __restrict__ fei1, int N) {
  int n = blockIdx.x * blockDim.x + threadIdx.x;
  if (n >= N) return;
  const float x = aff[n * 12 + 3];
  const float y = aff[n * 12 + 7];
  const float z = aff[n * 12 + 11];
  pos_out[n * 3 + 0] = x;
  pos_out[n * 3 + 1] = y;
  pos_out[n * 3 + 2] = z;
  P4[n] = make_float4(x, y, z, 0.0f);
  rsq[n] = x * x + y * y + z * z;
#pragma unroll
  for (int c = 0; c < 3; ++c) {
    const float p = (c == 0) ? x : ((c == 1) ? y : z);
#pragma unroll
    for (int t = 0; t < 8; ++t) {
      const float a = p * kFreq[t];
      emb_out[n * 48 + c * PED + t]     = __sinf(a);
      emb_out[n * 48 + c * PED + 8 + t] = __cosf(a);
    }
  }
#pragma unroll
  for (int k = 0; k < NUM_N; ++k) fei1[n * NUM_N + k] = n;
}

// ---------------------------------------------------------------------------
// Kernel B: WMMA-tiled kNN. Block = 128 threads (4 wave32), owns 16 rows.
// Wave w handles column tiles jt = w, w+4, ... Per tile:
//   A (16x4 f32): lanes 0-15 rows M=0..15 K={0,1}; lanes 16-31 K={2,3}
//   B (4x16 f32): mirrored layout for columns
//   D (16x16 f32): vgpr v, lane l -> (M = v + 8*(l>>4), N = l&15)
// Selection: 2 lanes per row; lane parity p owns contiguous columns 8p..8p+7
// so its 8 candidates per tile are two aligned ds_load_b128.
// ---------------------------------------------------------------------------
__global__ __launch_bounds__(128) void knn_kernel(
    const float4* __restrict__ P4, const float* __restrict__ rsq,
    int* __restrict__ ei, int* __restrict__ fei0, int N) {
  const int lane = threadIdx.x & 31;
  const int wave = threadIdx.x >> 5;
  const int hi   = lane >> 4;
  const int nl   = lane & 15;
  const int rowBase = blockIdx.x * 16;

  // Row stride 20 floats (80 B): keeps &sD[w][r][8p] 16-byte aligned for b128
  // reads and the [M][nl] write pattern bank-conflict-free (20 dwords mod 64).
  __shared__ float sD[4][16][20];          // per-wave d^2 tile staging
  __shared__ float sCd[16][8][NUM_N];      // per (row, wave*2+parity) candidates
  __shared__ int   sCi[16][8][NUM_N];
  __shared__ float sFd[16][2][NUM_N];      // final pair-merge staging
  __shared__ int   sFi[16][2][NUM_N];

  // A tile (invariant over the column loop).
  int arow = rowBase + nl; if (arow > N - 1) arow = N - 1;
  const float4 pa = P4[arow];
  v2f A;
  A.x = hi ? pa.z : pa.x;   // K = 2*hi
  A.y = hi ? 0.0f : pa.y;   // K = 2*hi + 1 (K=3 zero pad)

  float ra[8];
#pragma unroll
  for (int v = 0; v < 8; ++v) {
    int r = rowBase + v + 8 * hi; if (r > N - 1) r = N - 1;
    ra[v] = rsq[r];
  }

  float bd[NUM_N]; int bi[NUM_N];
#pragma unroll
  for (int k = 0; k < NUM_N; ++k) { bd[k] = __builtin_inff(); bi[k] = -1; }

  const int numTiles = (N + 15) >> 4;
  const int selRow = lane >> 1;   // 2 lanes per row for selection
  const int selPar = lane & 1;

  for (int jt = wave; jt < numTiles; jt += 4) {
    const int colBase = jt << 4;
    const int gj = colBase + nl;
    int jc = gj; if (jc > N - 1) jc = N - 1;
    const float4 pb = P4[jc];

    // Prefetch this wave's next column tile (emits global_prefetch_b8).
    int pf = colBase + 64 + nl; if (pf > N - 1) pf = N - 1;
    __builtin_prefetch((const void*)(P4 + pf), 0, 1);

    v2f B;
    B.x = hi ? pb.z : pb.x;
    B.y = hi ? 0.0f : pb.y;
    const float rb = pb.x * pb.x + pb.y * pb.y + pb.z * pb.z;

    v8f acc = {0.f, 0.f, 0.f, 0.f, 0.f, 0.f, 0.f, 0.f};
    acc = __builtin_amdgcn_wmma_f32_16x16x4_f32(
        /*neg_a=*/false, A, /*neg_b=*/false, B,
        /*c_mod=*/(short)0, acc, /*reuse_a=*/false, /*reuse_b=*/false);

#pragma unroll
    for (int v = 0; v < 8; ++v) {
      const int M = v + 8 * hi;
      float d2 = ra[v] + rb - 2.0f * acc[v];
      const int gi = rowBase + M;
      if (gi == gj || gj >= N || gi >= N) d2 = __builtin_inff();
      sD[wave][M][nl] = d2;
    }
    // Same-wave LDS RAW: DS ops are in-order per wave; wait + compiler barrier.
    asm volatile("s_wait_dscnt 0" ::: "memory");

    // Two aligned b128 loads give this lane's 8 contiguous candidates.
    {
      const float* rowp = &sD[wave][selRow][8 * selPar];
      const float4 c0 = *(const float4*)(rowp);
      const float4 c1 = *(const float4*)(rowp + 4);
      const int jb = colBase + 8 * selPar;
      insert20(bd, bi, c0.x, jb + 0);
      insert20(bd, bi, c0.y, jb + 1);
      insert20(bd, bi, c0.z, jb + 2);
      insert20(bd, bi, c0.w, jb + 3);
      insert20(bd, bi, c1.x, jb + 4);
      insert20(bd, bi, c1.y, jb + 5);
      insert20(bd, bi, c1.z, jb + 6);
      insert20(bd, bi, c1.w, jb + 7);
    }
    asm volatile("s_wait_dscnt 0" ::: "memory");  // WAR before next tile's store
  }

  // Publish each lane's local top-20 (register arrays -> must fully unroll).
#pragma unroll
  for (int k = 0; k < NUM_N; ++k) {
    sCd[selRow][wave * 2 + selPar][k] = bd[k];
    sCi[selRow][wave * 2 + selPar][k] = bi[k];
  }
  __syncthreads();

  if (wave == 0) {
    float fd[NUM_N]; int fi[NUM_N];
#pragma unroll
    for (int k = 0; k < NUM_N; ++k) { fd[k] = __builtin_inff(); fi[k] = -1; }

    // 160 candidates per row; each selection lane merges a contiguous half.
#pragma unroll 1
    for (int slot = 4 * selPar; slot < 4 * selPar + 4; ++slot) {
#pragma unroll 1
      for (int k = 0; k < NUM_N; ++k)
        insert20(fd, fi, sCd[selRow][slot][k], sCi[selRow][slot][k]);
    }

#pragma unroll
    for (int k = 0; k < NUM_N; ++k) {
      sFd[selRow][selPar][k] = fd[k];
      sFi[selRow][selPar][k] = fi[k];
    }
    asm volatile("s_wait_dscnt 0" ::: "memory");

    if (selPar == 0) {
#pragma unroll 1
      for (int k = 0; k < NUM_N; ++k)
        insert20(fd, fi, sFd[selRow][1][k], sFi[selRow][1][k]);
      const int row = rowBase + selRow;
      if (row < N) {
#pragma unroll
        for (int k = 0; k < NUM_N; ++k) {
          ei[row * NUM_N + k]   = fi[k];   // edge_index (ascending d^2)
          fei0[row * NUM_N + k] = fi[k];   // full_edge_index row 0
        }
      }
    }
  }
}

// ---------------------------------------------------------------------------
// Kernel C: per-edge rotation (rot^T * rel) + 16-dim distance encoding.
// ---------------------------------------------------------------------------
__global__ void edge_kernel(const float* __restrict__ aff,
                            const float4* __restrict__ P4,
                            const int* __restrict__ ei,
                            float* __restrict__ npos,
                            float* __restrict__ ndist, int N) {
  const int e = blockIdx.x * blockDim.x + threadIdx.x;
  if (e >= N * NUM_N) return;
  const int i = e / NUM_N;
  const int j = ei[e];
  const float4 pi4 = P4[i];
  const float4 pj4 = P4[j];
  const float rx = pj4.x - pi4.x, ry = pj4.y - pi4.y, rz = pj4.z - pi4.z;
  // out[c] = sum_jr rot[jr][c] * rel[jr];  rot[jr][c] = aff[i*12 + jr*4 + c]
  const float o0 = aff[i*12+0]*rx + aff[i*12+4]*ry + aff[i*12+8]*rz;
  const float o1 = aff[i*12+1]*rx + aff[i*12+5]*ry + aff[i*12+9]*rz;
  const float o2 = aff[i*12+2]*rx + aff[i*12+6]*ry + aff[i*12+10]*rz;
  npos[e * 3 + 0] = o0;
  npos[e * 3 + 1] = o1;
  npos[e * 3 + 2] = o2;
  const float d = sqrtf(o0 * o0 + o1 * o1 + o2 * o2);
#pragma unroll
  for (int t = 0; t < 8; ++t) {
    const float a = d * kFreq[t];
    ndist[e * PED + t]     = __sinf(a);
    ndist[e * PED + 8 + t] = __cosf(a);
  }
}

// ---------------------------------------------------------------------------
// Host launcher. Output layout (flat concat, reference return order):
//   pos3d_emb[N*48] | positions[N*3] | neighbour_positions[N*60] |
//   neighbour_distances[N*320] | edge_index[N*20 int] | full_edge_index[N*40 int]
// Workspace: float4 P4[N] (16B each) then float rsq[N]  (~200 KB for N=10000).
// ---------------------------------------------------------------------------
extern "C" void kernel_launch(void* const* d_in, const int* in_sizes, int n_in,
                              void* d_out, int out_size, void* d_ws, size_t ws_size,
                              hipStream_t stream) {
  const int N = in_sizes[0] / 12;
  const float* aff = (const float*)d_in[0];

  float* out   = (float*)d_out;
  float* pos3d = out;
  float* pos   = pos3d + (size_t)N * 48;
  float* npos  = pos   + (size_t)N * 3;
  float* ndist = npos  + (size_t)N * 60;
  int*   ei    = (int*)(ndist + (size_t)N * 320);
  int*   fei0  = ei   + (size_t)N * NUM_N;
  int*   fei1  = fei0 + (size_t)N * NUM_N;

  float4* P4  = (float4*)d_ws;
  float*  rsq = (float*)((char*)d_ws + (((size_t)N * 16 + 255) & ~(size_t)255));

  prep_kernel<<<(N + 255) / 256, 256, 0, stream>>>(aff, pos, pos3d, P4, rsq, fei1, N);
  knn_kernel<<<(N + 15) / 16, 128, 0, stream>>>(P4, rsq, ei, fei0, N);
  edge_kernel<<<(N * NUM_N + 255) / 256, 256, 0, stream>>>(aff, P4, ei, npos, ndist, N);
}